// P4DConv_695784702574
// MI455X (gfx1250) — compile-verified
//
#include <hip/hip_runtime.h>

typedef __attribute__((ext_vector_type(2))) float v2f;
typedef __attribute__((ext_vector_type(8))) float v8f;

#define NPTS   2048
#define MANCH  64
#define KNBR   32
#define COUT   1024
#define TDIM   16
#define R2CUT  0.49f

// ---------------------------------------------------------------------------
// Kernel 1: furthest point sampling, one block per (b,t) cloud.
// Exactly mirrors the reference scan: idxs[i] = last; d2 = min(d2, ||p-p_last||^2);
// last = first-argmax(d2). Writes anchor coords to workspace and new_xyzs output.
// ---------------------------------------------------------------------------
__global__ __launch_bounds__(256) void fps_kernel(const float* __restrict__ xyzs,
                                                  float* __restrict__ anchors,
                                                  float* __restrict__ new_xyzs) {
    __shared__ float s_d2[NPTS];
    __shared__ float s_rv[256];
    __shared__ int   s_ri[256];
    __shared__ int   s_last;

    const int tid   = threadIdx.x;
    const int cloud = blockIdx.x;                       // = bt
    const float* P  = xyzs + (size_t)cloud * NPTS * 3;

    for (int k = 0; k < NPTS / 256; ++k) s_d2[tid + k * 256] = 1e10f;
    if (tid == 0) s_last = 0;
    __syncthreads();

    for (int it = 0; it < MANCH; ++it) {
        const int last = s_last;
        const float cx = P[last * 3 + 0];
        const float cy = P[last * 3 + 1];
        const float cz = P[last * 3 + 2];
        if (tid == 0) {
            const size_t o = ((size_t)cloud * MANCH + it) * 3;
            anchors[o + 0] = cx; anchors[o + 1] = cy; anchors[o + 2] = cz;
            new_xyzs[o + 0] = cx; new_xyzs[o + 1] = cy; new_xyzs[o + 2] = cz;
        }
        float bv = -1.0f;
        int   bi = 0x7fffffff;
        for (int k = 0; k < NPTS / 256; ++k) {
            const int j = tid + k * 256;
            const float dx = P[j * 3 + 0] - cx;
            const float dy = P[j * 3 + 1] - cy;
            const float dz = P[j * 3 + 2] - cz;
            const float d  = dx * dx + dy * dy + dz * dz;
            const float nd = fminf(s_d2[j], d);
            s_d2[j] = nd;
            if (nd > bv || (nd == bv && j < bi)) { bv = nd; bi = j; }
        }
        s_rv[tid] = bv; s_ri[tid] = bi;
        __syncthreads();
        for (int s = 128; s > 0; s >>= 1) {
            if (tid < s) {
                const float ov = s_rv[tid + s];
                const int   oi = s_ri[tid + s];
                if (ov > s_rv[tid] || (ov == s_rv[tid] && oi < s_ri[tid])) {
                    s_rv[tid] = ov; s_ri[tid] = oi;
                }
            }
            __syncthreads();
        }
        if (tid == 0) s_last = s_ri[0];
        __syncthreads();
    }
}

// ---------------------------------------------------------------------------
// Kernel 2: ball query + grouped displacement + 1x1 conv (f32 WMMA 16x16x4)
// + ReLU + max-over-K + sum-over-dt. One wave32 per (bt, m) anchor.
// ---------------------------------------------------------------------------
__global__ __launch_bounds__(128) void p4d_feat_kernel(const float* __restrict__ xyzs,
                                                       const float* __restrict__ w_d,
                                                       const float* __restrict__ anchors,
                                                       float* __restrict__ out_feat) {
    const int lane = threadIdx.x & 31;
    const int wv   = threadIdx.x >> 5;
    const int wid  = blockIdx.x * 4 + wv;               // global wave id: [0, 4096)
    const int bt   = wid >> 6;
    const int m    = wid & 63;
    const int b    = bt >> 4;
    const int t    = bt & 15;

    __shared__ float sd[4][KNBR][3];                    // per-wave compaction buffer

    const float ax = anchors[wid * 3 + 0];
    const float ay = anchors[wid * 3 + 1];
    const float az = anchors[wid * 3 + 2];

    // A-matrix register tiles for the 3 temporal taps, ISA 16x4 f32 layout:
    // lane L holds row M = L%16; lanes 0-15 -> (K0,K1), lanes 16-31 -> (K2,K3).
    v2f A0[3], A1[3];

    for (int di = 0; di < 3; ++di) {
        const int dt = di - 1;
        int ts = t + dt;
        ts = ts < 0 ? 0 : (ts > TDIM - 1 ? TDIM - 1 : ts);
        const float* P = xyzs + (size_t)(b * TDIM + ts) * NPTS * 3;

        // ball query: first KNBR in-index-order points with d2 < R^2
        int have = 0;
        for (int c = 0; c < NPTS / 32 && have < KNBR; ++c) {
            const int j = c * 32 + lane;
            const float dx = P[j * 3 + 0] - ax;
            const float dy = P[j * 3 + 1] - ay;
            const float dz = P[j * 3 + 2] - az;
            const float d2 = dx * dx + dy * dy + dz * dz;
            const bool pred = d2 < R2CUT;
            const unsigned mask = (unsigned)__ballot(pred);
            const int pos = have + __popc(mask & ((1u << lane) - 1u));
            if (pred && pos < KNBR) {
                sd[wv][pos][0] = dx; sd[wv][pos][1] = dy; sd[wv][pos][2] = dz;
            }
            have += __popc(mask);
        }
        __builtin_amdgcn_wave_barrier();   // order wave-private LDS writes vs reads

        const int nv = have < KNBR ? have : KNBR;
        float ddx, ddy, ddz;
        if (lane < nv) {
            ddx = sd[wv][lane][0]; ddy = sd[wv][lane][1]; ddz = sd[wv][lane][2];
        } else if (have > 0) {             // pad with first valid neighbor
            ddx = sd[wv][0][0]; ddy = sd[wv][0][1]; ddz = sd[wv][0][2];
        } else {                           // no valid neighbor -> index 0
            ddx = P[0] - ax; ddy = P[1] - ay; ddz = P[2] - az;
        }
        const float dtf = (float)dt;

        // assemble A tiles via shuffles (lane l owns neighbor slot l)
        const int r = lane & 15;
        const float sx0 = __shfl(ddx, r, 32),      sy0 = __shfl(ddy, r, 32),      sz0 = __shfl(ddz, r, 32);
        const float sx1 = __shfl(ddx, r + 16, 32), sy1 = __shfl(ddy, r + 16, 32), sz1 = __shfl(ddz, r + 16, 32);
        const bool lo = lane < 16;
        v2f a0, a1;
        a0.x = lo ? sx0 : sz0;  a0.y = lo ? sy0 : dtf;
        a1.x = lo ? sx1 : sz1;  a1.y = lo ? sy1 : dtf;
        A0[di] = a0;
        A1[di] = a1;
    }

    // 64 column tiles of 16 output channels; B = W^T slice in 4x16 f32 layout:
    // lane L holds column N = L%16; lanes 0-15 -> rows (K0,K1), 16-31 -> (K2,K3).
    float* outp = out_feat + (size_t)bt * COUT * MANCH + m;
    const int cidx = lane & 15;
    const int k0   = (lane < 16) ? 0 : 2;

    for (int ct = 0; ct < COUT / 16; ++ct) {
        const int c = ct * 16 + cidx;
        v2f Bv;
        Bv.x = w_d[c * 4 + k0];
        Bv.y = w_d[c * 4 + k0 + 1];

        float acc = 0.0f;
        for (int di = 0; di < 3; ++di) {
            v8f z = {};
            v8f D0 = __builtin_amdgcn_wmma_f32_16x16x4_f32(
                false, A0[di], false, Bv, (short)0, z, false, false);
            v8f D1 = __builtin_amdgcn_wmma_f32_16x16x4_f32(
                false, A1[di], false, Bv, (short)0, z, false, false);
            // max over K=32 rows: 16 rows live in this lane's 8+8 VGPRs,
            // the other 16 in lane^16 (same column N = lane%16).
            float mx = D0[0];
            mx = fmaxf(mx, D0[1]); mx = fmaxf(mx, D0[2]); mx = fmaxf(mx, D0[3]);
            mx = fmaxf(mx, D0[4]); mx = fmaxf(mx, D0[5]); mx = fmaxf(mx, D0[6]); mx = fmaxf(mx, D0[7]);
            mx = fmaxf(mx, D1[0]); mx = fmaxf(mx, D1[1]); mx = fmaxf(mx, D1[2]); mx = fmaxf(mx, D1[3]);
            mx = fmaxf(mx, D1[4]); mx = fmaxf(mx, D1[5]); mx = fmaxf(mx, D1[6]); mx = fmaxf(mx, D1[7]);
            mx = fmaxf(mx, __shfl_xor(mx, 16, 32));
            acc += fmaxf(mx, 0.0f);        // ReLU after max (ReLU is monotone)
        }
        if (lane < 16) outp[(size_t)c * MANCH] = acc;
    }
}

// ---------------------------------------------------------------------------
extern "C" void kernel_launch(void* const* d_in, const int* in_sizes, int n_in,
                              void* d_out, int out_size, void* d_ws, size_t ws_size,
                              hipStream_t stream) {
    const float* xyzs = (const float*)d_in[0];   // [4,16,2048,3] f32
    const float* w_d  = (const float*)d_in[1];   // [1024,4] f32
    float* out        = (float*)d_out;           // new_xyzs (12288) at 0, new_features (4194304) at offset 196608
    float* anchors    = (float*)d_ws;            // [64,64,3] f32 scratch

    fps_kernel<<<64, 256, 0, stream>>>(xyzs, anchors, out);
    p4d_feat_kernel<<<1024, 128, 0, stream>>>(xyzs, w_d, anchors, out + 64 * TDIM * MANCH * 3 / TDIM * TDIM);
}